// ViterbiDecoder_8804682957173
// MI455X (gfx1250) — compile-verified
//
#include <hip/hip_runtime.h>
#include <stdint.h>

#define NN 2048
#define MM 2048
#define NP1 2049
#define MP1 2049
#define DMAXV (NN + MM - 2)   // 4094 : last diagonal d = i + j
#define NTHREADS 1024

typedef uint32_t u32;
typedef uint64_t u64;
typedef __attribute__((ext_vector_type(4))) u32 vu32x4;
typedef __attribute__((ext_vector_type(8))) u32 vu32x8;

// ---------------------------------------------------------------- memset ----
__global__ void zero_kernel(float* __restrict__ p, int n) {
  int i = blockIdx.x * blockDim.x + threadIdx.x;
  int st = gridDim.x * blockDim.x;
  for (; i < n; i += st) p[i] = 0.0f;
}

// ------------------------------------------------------------ TDM helpers ---
template <int NCNT>
__device__ __forceinline__ void wait_tensorcnt() {
#if __has_builtin(__builtin_amdgcn_s_wait_tensorcnt)
  __builtin_amdgcn_s_wait_tensorcnt(NCNT);
#else
  asm volatile("s_wait_tensorcnt %0" ::"i"(NCNT));
#endif
}

// One TDM descriptor: gather `len` floats with a stride of (MM-1) elements
// (theta anti-diagonal) into contiguous LDS at byte offset `lds_off`.
// D# layout per CDNA5 ISA ch.8: group0 (4 SGPRs) + group1 (8 SGPRs), 2D tile:
//   tile_dim0 = 1, tile_dim1 = len, tensor_dim0_stride = MM-1, data_size = 4B.
__device__ __forceinline__ void tdm_load_diag(u32 lds_off, u64 gaddr, u32 len) {
  vu32x4 g0;
  g0.x = 1u;                                  // count=1, no gather, user mode
  g0.y = lds_off;                             // lds_addr  (bits 63:32)
  g0.z = (u32)gaddr;                          // global_addr[31:0]
  g0.w = (u32)(gaddr >> 32) | (2u << 30);     // global_addr[56:32] | type=2
  vu32x8 g1;
  g1.s0 = (2u << 16);                         // wg_mask=0, data_size=2 (4B)
  g1.s1 = (1u << 16);                         // tensor_dim0[15:0] = 1
  g1.s2 = (len << 16);                        // dim0 hi=0 | tensor_dim1[15:0]
  g1.s3 = (1u << 16);                         // dim1 hi=0 | tile_dim0 = 1
  g1.s4 = len;                                // tile_dim1 = len, tile_dim2 = 0
  g1.s5 = (u32)(MM - 1);                      // tensor_dim0_stride[31:0]
  g1.s6 = 0u;                                 // stride hi, dim1_stride lo
  g1.s7 = 0u;
  asm volatile("tensor_load_to_lds %0, %1" ::"s"(g0), "s"(g1) : "memory");
}

// --------------------------------------------------------------- math -------
__device__ __forceinline__ float lse3(float a, float b, float c) {
  float m = fmaxf(a, fmaxf(b, c));
  return m + __logf(__expf(a - m) + __expf(b - m) + __expf(c - m));
}

// ------------------------------------------------------------ DP kernel -----
__global__ __launch_bounds__(NTHREADS) void viterbi_wavefront(
    const float* __restrict__ theta, const float* __restrict__ psi,
    const float* __restrict__ phi, const float* __restrict__ A,
    float* __restrict__ out) {
  __shared__ float vbuf[3][NP1 * 3];   // rotating V diagonals (74 KB)
  __shared__ float thbuf[3][NN];       // rotating theta diagonals (24 KB)

  const int tid = threadIdx.x;

  const float delta = A[0];
  const float epsv  = A[1];
  const float l12d = logf(1.0f - 2.0f * delta + 1e-10f);
  const float le1  = logf(1.0f - epsv + 1e-10f);
  const float ldl  = logf(delta + 1e-10f);
  const float lel  = logf(epsv + 1e-10f);

  // Each thread owns rows tid and tid+1024 for the whole run -> psi/phi are
  // loop-invariant per thread (note: reference indexes BOTH psi and phi by i).
  const int ia = tid;
  const int ib = tid + NTHREADS;
  const float psa = psi[ia], pha = phi[ia];
  const float psb = psi[ib], phb = phi[ib];

  // Zero all rotating V buffers (boundary-as-zero initial condition).
  for (int k = tid; k < 3 * NP1 * 3; k += NTHREADS) (&vbuf[0][0])[k] = 0.0f;

  auto issue = [&](int dd) {
    int j0 = dd - (MM - 1);
    int q0 = j0 > 1 ? j0 : 1;                       // i0 = max(1, d-(M-1))
    int q1 = (dd - 1 < NN - 1) ? (dd - 1) : (NN - 1);  // i1 = min(N-1, d-1)
    u32 len = (u32)(q1 - q0 + 1);
    // element offset of theta[i0, d-i0] = d + i0*(M-1)
    u64 ga = (u64)(uintptr_t)theta + 4ull * (u64)(dd + q0 * (MM - 1));
    u32 lo = (u32)(uintptr_t)(&thbuf[dd % 3][0]);   // LDS byte offset
    tdm_load_diag(lo, ga, len);
  };

  // Prologue: prefetch diagonals 2 and 3; guarantee diag 2 is resident.
  if (tid == 0) {
    issue(2);
    issue(3);
    wait_tensorcnt<1>();   // oldest (d=2) complete before first barrier
  }

  for (int d = 2; d <= DMAXV; ++d) {
    __syncthreads();  // publishes V(d-1), V(d-2) and theta(d) to all waves
    if (tid == 0) {
      if (d + 2 <= DMAXV) { issue(d + 2); wait_tensorcnt<1>(); }  // d+1 done
      else                { wait_tensorcnt<0>(); }                // drain tail
    }
    const int i0 = (d - (MM - 1) > 1) ? (d - (MM - 1)) : 1;
    const int i1 = (d - 1 < NN - 1) ? (d - 1) : (NN - 1);
    const int bc = d % 3;        // write V(d)
    const int b1 = (d + 2) % 3;  // == (d-1) % 3  : V[i-1,j], V[i,j-1]
    const int b2 = (d + 1) % 3;  // == (d-2) % 3  : V[i-1,j-1]

    auto row = [&](int i, float ps, float ph) {
      if (i >= i0 && i <= i1) {
        const float th = thbuf[bc][i - i0];
        const float* Vd = &vbuf[b2][(i - 1) * 3];
        const float* Vu = &vbuf[b1][(i - 1) * 3];
        const float* Vl = &vbuf[b1][i * 3];
        const float vm = lse3(th + l12d + Vd[0], th + le1 + Vd[1],
                              th + le1 + Vd[2]);
        const float vx = lse3(ps + ldl + Vu[0], ps + lel + Vu[1], 0.0f);
        const float vy = lse3(ph + ldl + Vl[0], 0.0f, ph + lel + Vl[2]);
        float* W = &vbuf[bc][i * 3];
        W[0] = vm; W[1] = vx; W[2] = vy;
        float* O = out + ((size_t)i * MP1 + (size_t)(d - i)) * 3;
        O[0] = vm; O[1] = vx; O[2] = vy;
      } else if (i == i0 - 1 || i == i1 + 1) {
        // zero fringe rows so neighbor reads on d+1/d+2 see the boundary 0
        float* W = &vbuf[bc][i * 3];
        W[0] = 0.0f; W[1] = 0.0f; W[2] = 0.0f;
      }
    };
    row(ia, psa, pha);
    row(ib, psb, phb);
    if (tid == 0 && i1 == NN - 1) {      // fringe row i = 2048
      float* W = &vbuf[bc][NN * 3];
      W[0] = 0.0f; W[1] = 0.0f; W[2] = 0.0f;
    }
  }
}

// --------------------------------------------------------------- launch -----
extern "C" void kernel_launch(void* const* d_in, const int* in_sizes, int n_in,
                              void* d_out, int out_size, void* d_ws,
                              size_t ws_size, hipStream_t stream) {
  (void)in_sizes; (void)n_in; (void)d_ws; (void)ws_size;
  const float* theta = (const float*)d_in[0];
  const float* psi   = (const float*)d_in[1];
  const float* phi   = (const float*)d_in[2];
  const float* A     = (const float*)d_in[3];
  float* out = (float*)d_out;

  // 1) boundary / invalid cells of V are zero -> zero-fill the whole output.
  zero_kernel<<<2048, 256, 0, stream>>>(out, out_size);
  // 2) single-WGP persistent wavefront: 32 wave32's, all state in LDS,
  //    TDM triple-buffered theta-diagonal gather two steps ahead.
  viterbi_wavefront<<<1, NTHREADS, 0, stream>>>(theta, psi, phi, A, out);
}